// SRLEmbeddings_40003325395092
// MI455X (gfx1250) — compile-verified
//
#include <hip/hip_runtime.h>
#include <stdint.h>

// Problem constants (match the reference)
static constexpr int kB = 16;
static constexpr int kS = 32;
static constexpr int kL = 128;   // sentence length (K dimension)
static constexpr int kD = 768;   // embedding dim (N dimension)
static constexpr int kA = 10;    // args per sentence
static constexpr int kT = 5;     // tokens per arg

static constexpr int kWRows   = 32;    // 1 sent row + 30 arg rows + 1 pad row
static constexpr int kWStride = 132;   // conflict-free ds_load_b64 A-fragment reads

static constexpr int kSlabK   = 16;    // K rows per LDS slab
static constexpr int kNSlabs  = kL / kSlabK;          // 8
static constexpr int kSlabStr = kD + 16;              // 784: 2*784 % 64 == 32 -> B frags conflict-free
static constexpr int kChunksPerWave = 12;             // 16 rows * 6 chunks / 8 waves

typedef float v2f __attribute__((ext_vector_type(2)));
typedef float v8f __attribute__((ext_vector_type(8)));

// --- CDNA5 async global->LDS copy (ASYNCcnt-tracked DMA, bypasses VGPRs) ---------
__device__ __forceinline__ uint32_t lds_offset(const void* p) {
    // generic shared pointer: addr[31:0] is the wave-relative LDS byte offset
    return (uint32_t)(uintptr_t)p;
}
__device__ __forceinline__ void async_copy_b128(uint32_t lds_byte_off, const float* g) {
    asm volatile("global_load_async_to_lds_b128 %0, %1, off"
                 :: "v"(lds_byte_off), "v"((unsigned long long)(uintptr_t)g)
                 : "memory");
}
#define WAIT_ASYNCCNT_12() asm volatile("s_wait_asynccnt 0xc" ::: "memory")
#define WAIT_ASYNCCNT_0()  asm volatile("s_wait_asynccnt 0x0" ::: "memory")

__global__ __launch_bounds__(256) void srl_wmma_kernel(
    const int*   __restrict__ sent_ids,   // [B,S,L]
    const float* __restrict__ sent_mask,  // [B,S,L]
    const float* __restrict__ emb,        // [B,S,L,D]
    const int*   __restrict__ pred_ids,   // [B,S,A,T]
    const int*   __restrict__ arg0_ids,   // [B,S,A,T]
    const int*   __restrict__ arg1_ids,   // [B,S,A,T]
    float* __restrict__ out_sent,         // [B,S,D]
    float* __restrict__ out_pred,         // [B,S,A,D]
    float* __restrict__ out_a0,           // [B,S,A,D]
    float* __restrict__ out_a1)           // [B,S,A,D]
{
    __shared__ __align__(16) float Eslab0[kSlabK * kSlabStr];  // 49.0 KB
    __shared__ __align__(16) float Eslab1[kSlabK * kSlabStr];  // 49.0 KB
    __shared__ float Wsh[kWRows * kWStride];                   // 16.5 KB
    __shared__ int   ls_ids[kL];
    __shared__ float ls_mask[kL];
    __shared__ int   toks[30 * kT];
    __shared__ int   cnts[30 * kT];
    __shared__ int   tok_sel[kWRows];
    __shared__ float invc[kWRows];
    __shared__ float msum;

    const int bs   = blockIdx.x;     // 0..B*S-1
    const int tid  = threadIdx.x;    // 0..255
    const int wave = tid >> 5;
    const int lane = tid & 31;
    const int half = lane >> 4;      // 0: lanes 0-15, 1: lanes 16-31
    const int lr   = lane & 15;

    const int*   sid = sent_ids  + (size_t)bs * kL;
    const float* msk = sent_mask + (size_t)bs * kL;
    const float* E   = emb       + (size_t)bs * kL * kD;

    // Kick off slab 0 DMA immediately: overlaps all of Phase 1.
    // Each wave moves 2 rows (6 x 512B chunks each) per slab.
    {
        #pragma unroll
        for (int j = 0; j < kChunksPerWave; ++j) {
            const int r = wave * 2 + (j >= 6 ? 1 : 0);
            const int p = (j >= 6) ? (j - 6) : j;
            const int e = p * 128 + lane * 4;
            async_copy_b128(lds_offset(&Eslab0[r * kSlabStr + e]),
                            E + (size_t)r * kD + e);
        }
    }

    // ---------------- Phase 1: build weight matrix W[32,128] in LDS ----------------
    if (tid == 0) msum = 0.f;
    if (tid < kL) { ls_ids[tid] = sid[tid]; ls_mask[tid] = msk[tid]; }
    __syncthreads();

    if (tid < kL) atomicAdd(&msum, ls_mask[tid]);   // ds_add_f32

    if (tid < 30 * kT) {   // one thread per (slot, t): count matches over L
        const int slot = tid / kT;            // 0-9 pred, 10-19 arg0, 20-29 arg1
        const int t    = tid % kT;
        const int r    = slot / kA;
        const int a    = slot % kA;
        const int* ap  = (r == 0 ? pred_ids : (r == 1 ? arg0_ids : arg1_ids))
                         + ((size_t)bs * kA + a) * kT;
        const int tok = ap[t];
        int c = 0;
        if (tok != 0) {
            #pragma unroll 16
            for (int l = 0; l < kL; ++l) c += (ls_ids[l] == tok) ? 1 : 0;
        }
        toks[tid] = tok;
        cnts[tid] = c;
    }
    __syncthreads();

    if (tid < kWRows) {    // pick LAST valid t per slot (reference overwrite semantics)
        int   sel = (int)0x80000000;   // sentinel: never equals a vocab id
        float ic  = 0.f;
        if (tid >= 1 && tid <= 30) {
            const int slot = tid - 1;
            #pragma unroll
            for (int t = kT - 1; t >= 0; --t) {
                const int c = cnts[slot * kT + t];
                if (c > 0) { sel = toks[slot * kT + t]; ic = 1.f / (float)c; break; }
            }
        }
        tok_sel[tid] = sel;
        invc[tid]    = ic;
    }
    __syncthreads();

    const float im = 1.f / fmaxf(msum, 1.f);
    for (int idx = tid; idx < kWRows * kL; idx += 256) {
        const int row = idx >> 7;
        const int l   = idx & (kL - 1);
        float w;
        if (row == 0)        w = ls_mask[l] * im;
        else if (row == 31)  w = 0.f;
        else                 w = (ls_ids[l] == tok_sel[row]) ? invc[row] : 0.f;
        Wsh[row * kWStride + l] = w;
    }

    // ------------- Phase 2: D[32,768] = W[32,128] x E[128,768], WMMA f32 -----------
    // Double-buffered async LDS slabs over K; 6 N-tiles x 2 M-tiles of accumulators
    // stay live across all 8 slabs.
    v8f acc[12];
    #pragma unroll
    for (int i = 0; i < 12; ++i) acc[i] = (v8f){0.f,0.f,0.f,0.f,0.f,0.f,0.f,0.f};

    for (int slab = 0; slab < kNSlabs; ++slab) {
        const float* cur = (slab & 1) ? Eslab1 : Eslab0;
        float*       nxt = (slab & 1) ? Eslab0 : Eslab1;

        if (slab + 1 < kNSlabs) {
            const float* Enext = E + (size_t)(slab + 1) * kSlabK * kD;
            #pragma unroll
            for (int j = 0; j < kChunksPerWave; ++j) {
                const int r = wave * 2 + (j >= 6 ? 1 : 0);
                const int p = (j >= 6) ? (j - 6) : j;
                const int e = p * 128 + lane * 4;
                async_copy_b128(lds_offset(&nxt[r * kSlabStr + e]),
                                Enext + (size_t)r * kD + e);
            }
            WAIT_ASYNCCNT_12();   // retire current slab's copies (next slab's 12 may fly)
        } else {
            WAIT_ASYNCCNT_0();
        }
        __syncthreads();          // all waves' copies of `cur` visible; W ready (slab 0)

        #pragma unroll
        for (int ntl = 0; ntl < 6; ++ntl) {
            const int nt   = wave + 8 * ntl;
            const int ncol = nt * 16 + lr;
            #pragma unroll
            for (int kl = 0; kl < kSlabK; kl += 4) {
                const int kk = kl + 2 * half;          // ISA A/B lane layout
                const int kg = slab * kSlabK + kk;     // global K index into W
                const v2f af0 = *(const v2f*)(&Wsh[lr * kWStride + kg]);
                const v2f af1 = *(const v2f*)(&Wsh[(lr + 16) * kWStride + kg]);
                v2f bf;
                bf.x = cur[kk * kSlabStr + ncol];
                bf.y = cur[(kk + 1) * kSlabStr + ncol];
                acc[2 * ntl]     = __builtin_amdgcn_wmma_f32_16x16x4_f32(
                                       false, af0, false, bf, (short)0, acc[2 * ntl], false, false);
                acc[2 * ntl + 1] = __builtin_amdgcn_wmma_f32_16x16x4_f32(
                                       false, af1, false, bf, (short)0, acc[2 * ntl + 1], false, false);
            }
        }
        __syncthreads();          // done reading `cur` before it is overwritten
    }

    // ---------------- Store: scatter D rows to the four outputs --------------------
    // C/D layout: M = v + 8*half (+16 for the second M-tile), N = lane & 15.
    const size_t bsD = (size_t)bs * kD;
    #pragma unroll
    for (int ntl = 0; ntl < 6; ++ntl) {
        const int ncol = (wave + 8 * ntl) * 16 + lr;
        #pragma unroll
        for (int v = 0; v < 8; ++v) {
            const int m0 = v + 8 * half;        // 0..15
            const int m1 = m0 + 16;             // 16..31
            const float val0 = acc[2 * ntl][v];
            const float val1 = acc[2 * ntl + 1][v];

            if (m0 == 0)
                out_sent[bsD + ncol] = val0;
            else if (m0 <= 10)
                out_pred[((size_t)bs * kA + (m0 - 1)) * kD + ncol] = val0;
            else  // 11..15 -> arg0 slots 0..4
                out_a0[((size_t)bs * kA + (m0 - 11)) * kD + ncol] = val0;

            if (m1 <= 20)       // 16..20 -> arg0 slots 5..9
                out_a0[((size_t)bs * kA + (m1 - 11)) * kD + ncol] = val1;
            else if (m1 <= 30)  // 21..30 -> arg1 slots 0..9
                out_a1[((size_t)bs * kA + (m1 - 21)) * kD + ncol] = val1;
            // m1 == 31: padding row, discard
        }
    }
}

extern "C" void kernel_launch(void* const* d_in, const int* in_sizes, int n_in,
                              void* d_out, int out_size, void* d_ws, size_t ws_size,
                              hipStream_t stream) {
    const int*   sid  = (const int*)  d_in[0];
    const float* msk  = (const float*)d_in[1];
    const float* emb  = (const float*)d_in[2];
    const int*   pred = (const int*)  d_in[3];
    const int*   a0   = (const int*)  d_in[4];
    const int*   a1   = (const int*)  d_in[5];

    float* out       = (float*)d_out;
    float* out_sent  = out;                                          // [B,S,D]
    float* out_pred  = out_sent + (size_t)kB * kS * kD;              // [B,S,A,D]
    float* out_a0    = out_pred + (size_t)kB * kS * kA * kD;
    float* out_a1    = out_a0   + (size_t)kB * kS * kA * kD;

    dim3 grid(kB * kS);   // 512 workgroups, one per (b,s)
    dim3 block(256);      // 8 wave32 per workgroup
    hipLaunchKernelGGL(srl_wmma_kernel, grid, block, 0, stream,
                       sid, msk, emb, pred, a0, a1,
                       out_sent, out_pred, out_a0, out_a1);
}